// MaskedRNNBase_37546604101892
// MI455X (gfx1250) — compile-verified
//
#include <hip/hip_runtime.h>
#include <hip/hip_bf16.h>
#include <math.h>

// ---------------------------------------------------------------------------
// 2-layer bidirectional masked LSTM for MI455X (gfx1250, wave32).
// fp32 numerics, matrix work on V_WMMA_F32_16X16X4_F32.
//  - Weights repacked once into (k,k+1)-pair layout -> every B fetch is one
//    aligned global_load_b64.
//  - Input projections: big WMMA GEMMs (M=T*B=16384, N=2048), LDS-staged A.
//  - Recurrent scans: persistent kernel, 32 blocks x 8 waves = the full
//    2x128 tile grid; block b owns gate columns [16b,16b+16) for all 4 gates,
//    so gate combine is block-local (LDS), c stays in registers, and a
//    double-buffered h needs only ONE device-wide barrier per timestep.
// ---------------------------------------------------------------------------

typedef __attribute__((ext_vector_type(2))) float v2f;
typedef __attribute__((ext_vector_type(8))) float v8f;

#define GN   2048            // 4*H gate width
#define HH   512             // hidden size
#define TT   512             // timesteps
#define BB   32              // batch
#define MROWS (TT*BB)        // 16384 rows for input-projection GEMMs
#define NBLK 32              // scan grid (co-resident)

// ---------------- workspace layout (float units) ---------------------------
constexpr size_t O_PIH0F = 0;                                  // packed Wih0 fwd
constexpr size_t O_PIH0B = O_PIH0F + (size_t)512 * GN;
constexpr size_t O_PHH0F = O_PIH0B + (size_t)512 * GN;
constexpr size_t O_PHH0B = O_PHH0F + (size_t)512 * GN;
constexpr size_t O_PIH1F = O_PHH0B + (size_t)512 * GN;
constexpr size_t O_PIH1B = O_PIH1F + (size_t)1024 * GN;
constexpr size_t O_PHH1F = O_PIH1B + (size_t)1024 * GN;
constexpr size_t O_PHH1B = O_PHH1F + (size_t)512 * GN;
constexpr size_t O_G     = O_PHH1B + (size_t)512 * GN;         // [16384, 2048]
constexpr size_t O_Y0    = O_G     + (size_t)MROWS * GN;       // [T,B,2H]
constexpr size_t O_H0    = O_Y0    + (size_t)TT * BB * 2 * HH; // h ping
constexpr size_t O_H1    = O_H0    + (size_t)BB * HH;          // h pong
constexpr size_t O_BAR   = O_H1    + (size_t)BB * HH;          // 2 ints

// ---------------- device-wide spin barrier ---------------------------------
__device__ __forceinline__ void gridBarrier(int* cnt, int* gen) {
    __syncthreads();
    if (threadIdx.x == 0) {
        __threadfence();
        int g = __hip_atomic_load(gen, __ATOMIC_RELAXED, __HIP_MEMORY_SCOPE_AGENT);
        if (atomicAdd(cnt, 1) == NBLK - 1) {
            __hip_atomic_store(cnt, 0, __ATOMIC_RELAXED, __HIP_MEMORY_SCOPE_AGENT);
            __hip_atomic_store(gen, g + 1, __ATOMIC_RELEASE, __HIP_MEMORY_SCOPE_AGENT);
        } else {
            while (__hip_atomic_load(gen, __ATOMIC_ACQUIRE, __HIP_MEMORY_SCOPE_AGENT) == g) { }
        }
        __threadfence();
    }
    __syncthreads();
}

__global__ void init_barrier_k(int* bar) { bar[0] = 0; bar[1] = 0; }

// ---------------- weight repack: W[2048,K] -> P[K/2][2048] of (k,k+1) pairs --
__global__ __launch_bounds__(256) void pack_k(const float* __restrict__ W,
                                              float* __restrict__ P, int K) {
    int idx = blockIdx.x * 256 + threadIdx.x;        // over (K/2)*2048
    int total = (K >> 1) * GN;
    if (idx < total) {
        int kk = idx >> 11;            // pair-row
        int n  = idx & (GN - 1);       // column
        P[((size_t)idx << 1)    ] = W[(size_t)n * K + 2 * kk];
        P[((size_t)idx << 1) + 1] = W[(size_t)n * K + 2 * kk + 1];
    }
}

// ---------------- input-projection GEMM: G = A[M,K] * W^T + b1 + b2 --------
// wave tile 16(M) x 64(N); A staged via padded LDS; packed-B b64 fetches.
__global__ __launch_bounds__(256) void gemm_bias_k(const float* __restrict__ A,
                                                   const float* __restrict__ Bpk,
                                                   const float* __restrict__ b1,
                                                   const float* __restrict__ b2,
                                                   float* __restrict__ G,
                                                   int K) {
    __shared__ float ash[16 * 130];
    const int tid  = threadIdx.x;
    const int wave = tid >> 5;
    const int lane = tid & 31;
    const int l16  = lane & 15;
    const int half = lane >> 4;
    const int wt    = blockIdx.x * 8 + wave;
    const int mtile = wt >> 5;
    const int n64   = wt & 31;
    const int row0  = mtile * 16;

    v8f zero = {};
    v8f acc[4] = {zero, zero, zero, zero};

    for (int kc = 0; kc < K; kc += 128) {
        for (int i = tid; i < 16 * 128; i += 256) {
            int r = i >> 7, c = i & 127;
            ash[r * 130 + c] = A[(size_t)(row0 + r) * K + kc + c];
        }
        __syncthreads();
        #pragma unroll 4
        for (int kk = 0; kk < 128; kk += 4) {
            v2f a = *(const v2f*)&ash[l16 * 130 + kk + 2 * half];
            const float* bp = Bpk + (((size_t)((kc + kk) >> 1) + half) << 12)
                                  + ((n64 * 64 + l16) << 1);
            #pragma unroll
            for (int nt = 0; nt < 4; ++nt) {
                v2f b = *(const v2f*)(bp + nt * 32);
                acc[nt] = __builtin_amdgcn_wmma_f32_16x16x4_f32(
                    false, a, false, b, (short)0, acc[nt], false, false);
            }
        }
        __syncthreads();
    }
    #pragma unroll
    for (int nt = 0; nt < 4; ++nt) {
        int col = n64 * 64 + nt * 16 + l16;
        float bsum = b1[col] + b2[col];
        #pragma unroll
        for (int j = 0; j < 8; ++j) {
            int r = row0 + j + 8 * half;
            G[(size_t)r * GN + col] = acc[nt][j] + bsum;
        }
    }
}

// ---------------- persistent masked-LSTM scan ------------------------------
// block b owns hidden cols [16b, 16b+16): wave w -> (gate = w>>1, m = w&1).
// One grid barrier per timestep; h double-buffered; c in registers.
__global__ __launch_bounds__(256) void lstm_scan_k(const float* __restrict__ G,    // [T*B, 2048]
                                                   const float* __restrict__ Bpk,  // packed Whh
                                                   const float* __restrict__ mask, // [T, B]
                                                   float* __restrict__ h0buf,      // [B*H] ping
                                                   float* __restrict__ h1buf,      // [B*H] pong
                                                   float* __restrict__ yout,       // [T,B,ystride]
                                                   int ycol0, int ystride,
                                                   float* __restrict__ hn,
                                                   float* __restrict__ cn,
                                                   int* bar, int reverse) {
    __shared__ float gsh[4][BB][16];               // 8 KB gate exchange
    const int tid  = threadIdx.x;
    const int wave = tid >> 5;
    const int lane = tid & 31;
    const int l16  = lane & 15;
    const int half = lane >> 4;
    const int gate = wave >> 1;                    // 0..3
    const int m    = wave & 1;                     // batch rows 0-15 / 16-31
    const int col  = gate * HH + blockIdx.x * 16 + l16;   // gate column 0..2047
    int* bcnt = bar;
    int* bgen = bar + 1;

    // this thread's two persistent state elements (b, j)
    const int i0  = tid * 2;
    const int b0  = i0 >> 4,        jj0 = i0 & 15;
    const int b1  = (i0 + 1) >> 4,  jj1 = (i0 + 1) & 15;
    const int j0  = blockIdx.x * 16 + jj0;
    const int j1  = blockIdx.x * 16 + jj1;
    float c0 = 0.0f, c1 = 0.0f;

    for (int i = blockIdx.x * 256 + tid; i < BB * HH; i += NBLK * 256) h0buf[i] = 0.0f;
    gridBarrier(bcnt, bgen);

    for (int s = 0; s < TT; ++s) {
        const int t = reverse ? (TT - 1 - s) : s;
        const float* hr = (s & 1) ? h1buf : h0buf;
        float*       hw = (s & 1) ? h0buf : h1buf;

        // phase 1: this wave's 16x16 tile of h @ Whh.T
        v8f acc = {};
        const float* arow = hr + (size_t)(m * 16 + l16) * HH;
        #pragma unroll 4
        for (int k = 0; k < HH; k += 4) {
            v2f a = *(const v2f*)(arow + k + 2 * half);
            v2f b = *(const v2f*)(Bpk + (((size_t)(k >> 1) + half) << 12) + (col << 1));
            acc = __builtin_amdgcn_wmma_f32_16x16x4_f32(
                false, a, false, b, (short)0, acc, false, false);
        }
        const float* Gt = G + (size_t)t * BB * GN;
        #pragma unroll
        for (int jr = 0; jr < 8; ++jr) {
            int br = m * 16 + jr + 8 * half;       // batch index 0..31
            gsh[gate][br][l16] = acc[jr] + Gt[(size_t)br * GN + col];
        }
        __syncthreads();

        // phase 2: block-local gate combine + masked state update
        const float mt0 = mask[t * BB + b0];
        const float mt1 = mask[t * BB + b1];
        {
            float ii = 1.0f / (1.0f + expf(-gsh[0][b0][jj0]));
            float ff = 1.0f / (1.0f + expf(-gsh[1][b0][jj0]));
            float g2 = tanhf(gsh[2][b0][jj0]);
            float oo = 1.0f / (1.0f + expf(-gsh[3][b0][jj0]));
            float hold = hr[(size_t)b0 * HH + j0];
            float cnv = ff * c0 + ii * g2;
            float hnv = oo * tanhf(cnv);
            hnv = hnv * mt0 + hold * (1.0f - mt0);
            cnv = cnv * mt0 + c0 * (1.0f - mt0);
            c0 = cnv;
            hw[(size_t)b0 * HH + j0] = hnv;
            yout[(size_t)(t * BB + b0) * ystride + ycol0 + j0] = hnv;
        }
        {
            float ii = 1.0f / (1.0f + expf(-gsh[0][b1][jj1]));
            float ff = 1.0f / (1.0f + expf(-gsh[1][b1][jj1]));
            float g2 = tanhf(gsh[2][b1][jj1]);
            float oo = 1.0f / (1.0f + expf(-gsh[3][b1][jj1]));
            float hold = hr[(size_t)b1 * HH + j1];
            float cnv = ff * c1 + ii * g2;
            float hnv = oo * tanhf(cnv);
            hnv = hnv * mt1 + hold * (1.0f - mt1);
            cnv = cnv * mt1 + c1 * (1.0f - mt1);
            c1 = cnv;
            hw[(size_t)b1 * HH + j1] = hnv;
            yout[(size_t)(t * BB + b1) * ystride + ycol0 + j1] = hnv;
        }
        gridBarrier(bcnt, bgen);   // also protects gsh reuse next step
    }

    // final state: T is even -> last write landed in h0buf
    hn[(size_t)b0 * HH + j0] = h0buf[(size_t)b0 * HH + j0];
    hn[(size_t)b1 * HH + j1] = h0buf[(size_t)b1 * HH + j1];
    cn[(size_t)b0 * HH + j0] = c0;
    cn[(size_t)b1 * HH + j1] = c1;
}

// ---------------------------------------------------------------------------
extern "C" void kernel_launch(void* const* d_in, const int* in_sizes, int n_in,
                              void* d_out, int out_size, void* d_ws, size_t ws_size,
                              hipStream_t stream) {
    const float* x    = (const float*)d_in[0];   // [T,B,512]
    const float* mask = (const float*)d_in[1];   // [T,B]
    const float* Wih0 = (const float*)d_in[2];   // [2,2048,512]
    const float* Whh0 = (const float*)d_in[3];   // [2,2048,512]
    const float* bih0 = (const float*)d_in[4];   // [2,2048]
    const float* bhh0 = (const float*)d_in[5];
    const float* Wih1 = (const float*)d_in[6];   // [2,2048,1024]
    const float* Whh1 = (const float*)d_in[7];   // [2,2048,512]
    const float* bih1 = (const float*)d_in[8];
    const float* bhh1 = (const float*)d_in[9];

    float* out    = (float*)d_out;
    float* hn_out = out + (size_t)TT * BB * 2 * HH;     // [4,B,H]
    float* cn_out = hn_out + (size_t)4 * BB * HH;       // [4,B,H]

    float* ws = (float*)d_ws;
    float* G  = ws + O_G;
    float* y0 = ws + O_Y0;
    float* h0 = ws + O_H0;
    float* h1 = ws + O_H1;
    int*  bar = (int*)(ws + O_BAR);

    init_barrier_k<<<1, 1, 0, stream>>>(bar);

    auto pk = [&](const float* W, float* P, int K) {
        int total = (K >> 1) * GN;
        pack_k<<<(total + 255) / 256, 256, 0, stream>>>(W, P, K);
    };
    pk(Wih0,                   ws + O_PIH0F, 512);
    pk(Wih0 + (size_t)GN*512,  ws + O_PIH0B, 512);
    pk(Whh0,                   ws + O_PHH0F, 512);
    pk(Whh0 + (size_t)GN*512,  ws + O_PHH0B, 512);
    pk(Wih1,                   ws + O_PIH1F, 1024);
    pk(Wih1 + (size_t)GN*1024, ws + O_PIH1B, 1024);
    pk(Whh1,                   ws + O_PHH1F, 512);
    pk(Whh1 + (size_t)GN*512,  ws + O_PHH1B, 512);

    const int gemm_blocks = (MROWS / 16) * (GN / 64) / 8;   // 4096

    // ---- layer 0 forward ----
    gemm_bias_k<<<gemm_blocks, 256, 0, stream>>>(x, ws + O_PIH0F, bih0, bhh0, G, 512);
    lstm_scan_k<<<NBLK, 256, 0, stream>>>(G, ws + O_PHH0F, mask, h0, h1,
                                          y0, 0, 2 * HH, hn_out, cn_out, bar, 0);
    // ---- layer 0 backward ----
    gemm_bias_k<<<gemm_blocks, 256, 0, stream>>>(x, ws + O_PIH0B, bih0 + GN, bhh0 + GN, G, 512);
    lstm_scan_k<<<NBLK, 256, 0, stream>>>(G, ws + O_PHH0B, mask, h0, h1,
                                          y0, HH, 2 * HH,
                                          hn_out + (size_t)BB * HH, cn_out + (size_t)BB * HH, bar, 1);
    // ---- layer 1 forward ----
    gemm_bias_k<<<gemm_blocks, 256, 0, stream>>>(y0, ws + O_PIH1F, bih1, bhh1, G, 1024);
    lstm_scan_k<<<NBLK, 256, 0, stream>>>(G, ws + O_PHH1F, mask, h0, h1,
                                          out, 0, 2 * HH,
                                          hn_out + (size_t)2 * BB * HH, cn_out + (size_t)2 * BB * HH, bar, 0);
    // ---- layer 1 backward ----
    gemm_bias_k<<<gemm_blocks, 256, 0, stream>>>(y0, ws + O_PIH1B, bih1 + GN, bhh1 + GN, G, 1024);
    lstm_scan_k<<<NBLK, 256, 0, stream>>>(G, ws + O_PHH1B, mask, h0, h1,
                                          out, HH, 2 * HH,
                                          hn_out + (size_t)3 * BB * HH, cn_out + (size_t)3 * BB * HH, bar, 1);
}